// SpatialGate_78254304133613
// MI455X (gfx1250) — compile-verified
//
#include <hip/hip_runtime.h>
#include <hip/hip_bf16.h>
#include <math.h>

typedef __attribute__((ext_vector_type(2))) float v2f;
typedef __attribute__((ext_vector_type(8))) float v8f;

#define NB    32
#define NC    256
#define HW    4096            // 64*64
#define PITCH 72              // padded LDS row pitch (floats)
#define ROWS  14              // 8-row band + 3 halo each side
#define PLANE (ROWS * PITCH)  // 1008 floats per channel plane

// ---------------------------------------------------------------------------
// Kernel 1: channel-wise max and mean over C=256.
// Each thread owns 2 consecutive pixels (float2 loads); a warp reads 256B
// contiguous per channel iteration -> fully coalesced. 256 blocks x 8 waves,
// channel loop unrolled x4 for load pipelining.
// ---------------------------------------------------------------------------
__global__ __launch_bounds__(256) void k_reduce(const float* __restrict__ x,
                                                float* __restrict__ comp) {
  int tid = blockIdx.x * 256 + threadIdx.x;      // 0..65535
  int b   = tid >> 11;                           // 2048 threads per batch
  int pp  = (tid << 1) & (HW - 1);
  const float* base = x + (size_t)b * NC * HW + pp;

  float mx0 = -__builtin_huge_valf(), mx1 = -__builtin_huge_valf();
  float sm0 = 0.f, sm1 = 0.f;
#pragma unroll 4
  for (int c = 0; c < NC; ++c) {
    float2 v = *(const float2*)(base + (size_t)c * HW);
    mx0 = fmaxf(mx0, v.x);
    mx1 = fmaxf(mx1, v.y);
    sm0 += v.x;
    sm1 += v.y;
  }
  float* cm = comp + (size_t)b * 2 * HW + pp;
  *(float2*)cm        = make_float2(mx0, mx1);
  *(float2*)(cm + HW) = make_float2(sm0 * (1.f / 256.f), sm1 * (1.f / 256.f));
}

// ---------------------------------------------------------------------------
// Kernel 2: 7x7 2->1 conv + sigmoid via V_WMMA_F32_16X16X4_F32.
// One block per (batch, 8-row band): 256 blocks. comp rows [r0-3, r0+11)
// staged into zero-padded LDS [3][14][72]; plane 2 stays zero so the k=98/99
// pad of the 98-long im2col vector reads zeros branch-free (EXEC all-ones at
// every WMMA). Per-k LDS byte offsets are precomputed into koff[100] so the
// inner iteration is: b64 offset-pair load, b64 weight-pair load, two b32
// A-gathers, one WMMA.
// Matvec trick: B[k][n] = w[k] for all n -> every output column carries the
// matvec result; lanes 0/16 hold pixels 0-7 / 8-15 in acc[0..7].
// ---------------------------------------------------------------------------
__global__ __launch_bounds__(256) void k_conv(const float* __restrict__ comp,
                                              const float* __restrict__ w7,
                                              float* __restrict__ scale) {
  __shared__ float s[3 * PLANE];   // 12096 B (plane 2 = zeros)
  __shared__ float wl[100];
  __shared__ int   koff[100];

  const int b    = blockIdx.x >> 3;
  const int band = blockIdx.x & 7;
  const int r0   = band << 3;                    // first output row of band
  const int tid  = threadIdx.x;

  for (int i = tid; i < 3 * PLANE; i += 256) s[i] = 0.f;
  if (tid < 100) {
    wl[tid] = (tid < 98) ? w7[tid] : 0.f;
    int c = tid / 49;                            // 2 for the pad ks
    int r = tid - c * 49;
    int dh = r / 7, dw = r - dh * 7;
    koff[tid] = (c * PLANE + dh * PITCH + dw) << 2;   // byte offset
  }
  __syncthreads();

  // stage comp rows [r0-3, r0+11) for both channels; out-of-image rows stay 0
  for (int i = tid; i < 2 * ROWS * 64; i += 256) {
    int c  = i / (ROWS * 64);
    int r  = i - c * (ROWS * 64);
    int lr = r >> 6;                             // LDS row 0..13
    int w  = r & 63;
    int gr = r0 + lr - 3;                        // global row
    if (gr >= 0 && gr < 64)
      s[c * PLANE + lr * PITCH + (w + 3)] =
          comp[((size_t)b * 2 + c) * HW + gr * 64 + w];
  }
  __syncthreads();

  const int wave = tid >> 5;
  const int lane = tid & 31;
  const int m    = lane & 15;                    // pixel within 16-wide group
  const int hi   = lane >> 4;                    // 0 -> K{0,1}, 1 -> K{2,3}
  const char* sb = (const char*)s;

  for (int g = wave; g < 32; g += 8) {           // 32 groups of 16 pixels
    const int hl = g >> 2;                       // local row 0..7
    const int wb = (g & 3) << 4;
    const int pbase = (hl * PITCH + wb + m) << 2;   // lane's patch base (bytes)
    v8f acc = {};
#pragma unroll 1
    for (int k0 = 0; k0 <= 96; k0 += 4) {
      const int kx = k0 + (hi << 1);             // even -> 8B aligned pairs
      int2   ko = *(const int2*)&koff[kx];
      float2 wv = *(const float2*)&wl[kx];
      v2f a, bv;
      a.x  = *(const float*)(sb + pbase + ko.x);
      a.y  = *(const float*)(sb + pbase + ko.y);
      bv.x = wv.x;
      bv.y = wv.y;
      acc = __builtin_amdgcn_wmma_f32_16x16x4_f32(
          /*neg_a=*/false, a, /*neg_b=*/false, bv,
          /*c_mod=*/(short)0, acc, /*reuse_a=*/false, /*reuse_b=*/false);
    }
    if (m == 0) {                                // lanes 0 and 16 hold results
      float* dst = scale + (size_t)b * HW + (r0 + hl) * 64 + wb + (hi << 3);
#pragma unroll
      for (int j = 0; j < 8; ++j) {
        float v = acc[j];
        dst[j] = __builtin_amdgcn_rcpf(1.f + __expf(-v));   // fast sigmoid
      }
    }
  }
}

// ---------------------------------------------------------------------------
// Kernel 3: out = x * scale, pure streaming at b128 granularity.
// 33.5M elements / 4 per thread / 256 per block = 32768 blocks.
// ---------------------------------------------------------------------------
__global__ __launch_bounds__(256) void k_scale(const float* __restrict__ x,
                                               const float* __restrict__ scale,
                                               float* __restrict__ out) {
  size_t f = ((size_t)blockIdx.x * 256 + threadIdx.x) * 4;
  size_t plane = f >> 12;              // b*256 + c
  int pix = (int)(f & (HW - 1));       // multiple of 4 -> aligned float4
  int b   = (int)(plane >> 8);

  float4 xv = *(const float4*)(x + f);
  float4 sv = *(const float4*)(scale + (size_t)b * HW + pix);
  float4 o;
  o.x = xv.x * sv.x;
  o.y = xv.y * sv.y;
  o.z = xv.z * sv.z;
  o.w = xv.w * sv.w;
  *(float4*)(out + f) = o;
}

extern "C" void kernel_launch(void* const* d_in, const int* in_sizes, int n_in,
                              void* d_out, int out_size, void* d_ws, size_t ws_size,
                              hipStream_t stream) {
  const float* x  = (const float*)d_in[0];   // [32,256,64,64] fp32
  const float* w7 = (const float*)d_in[1];   // [1,2,7,7] = 98 floats
  float* out = (float*)d_out;

  float* comp  = (float*)d_ws;               // [32,2,4096] = 1 MB
  float* scale = comp + (size_t)NB * 2 * HW; // [32,4096]   = 0.5 MB

  k_reduce<<<256, 256, 0, stream>>>(x, comp);
  k_conv<<<NB * 8, 256, 0, stream>>>(comp, w7, scale);
  k_scale<<<32768, 256, 0, stream>>>(x, scale, out);
}